// TimeAwareRandomWalkDiffusion_46858093199624
// MI455X (gfx1250) — compile-verified
//
#include <hip/hip_runtime.h>
#include <hip/hip_bf16.h>
#include <stdint.h>

#define ALPHA_F 0.1f

#ifndef __has_builtin
#define __has_builtin(x) 0
#endif

// CDNA5 async global->LDS streaming path (ASYNCcnt). Guarded so the file
// compiles on toolchains without the builtins (falls back to plain B128 loads).
#if defined(__HIP_DEVICE_COMPILE__) && \
    __has_builtin(__builtin_amdgcn_global_load_async_to_lds_b128) && \
    __has_builtin(__builtin_amdgcn_s_wait_asynccnt)
#define HAVE_ASYNC_LDS 1
#else
#define HAVE_ASYNC_LDS 0
#endif

#if HAVE_ASYNC_LDS
typedef int twrd_v4i __attribute__((ext_vector_type(4)));
typedef __attribute__((address_space(1))) twrd_v4i* twrd_g4p;  // global (AS1)
typedef __attribute__((address_space(3))) twrd_v4i* twrd_l4p;  // LDS (AS3)
__device__ __forceinline__ void twrd_async_cp16(const float4* g, float4* l) {
  __builtin_amdgcn_global_load_async_to_lds_b128((twrd_g4p)g, (twrd_l4p)l, 0, 0);
}
#endif

// ---------------------------------------------------------------------------
// Kernel 0: init scratch: maxbits = 0, deg[0..N) = 0   (N read from device)
// ---------------------------------------------------------------------------
__global__ __launch_bounds__(256) void twrd_init_kernel(unsigned* __restrict__ maxbits,
                                                        float* __restrict__ deg,
                                                        const int* __restrict__ nn) {
  long long g  = (long long)blockIdx.x * blockDim.x + threadIdx.x;
  long long gs = (long long)gridDim.x * blockDim.x;
  if (g == 0) *maxbits = 0u;
  long long N = *nn;
  for (long long i = g; i < N; i += gs) deg[i] = 0.0f;
}

// ---------------------------------------------------------------------------
// Kernel 1: T = max(edge_time).  Double-buffered async global->LDS streaming,
// wave32 shuffle reduction, one u32-bit atomicMax per block (times >= 0 so
// float bit pattern is monotonic).
// ---------------------------------------------------------------------------
__global__ __launch_bounds__(256) void twrd_max_kernel(const float* __restrict__ t,
                                                       unsigned* __restrict__ maxbits,
                                                       long long C,   // E/4 float4 elems
                                                       long long E) {
  const float4* t4 = (const float4*)t;
  const int tid = threadIdx.x;
  float m = 0.0f;
  const long long nChunks = (C + 255) >> 8;  // 256 float4 per block-chunk

#if HAVE_ASYNC_LDS
  __shared__ float4 buf[2][256];
  long long c = blockIdx.x;
  int p = 0;
  if (c < nChunks) {
    long long e0 = c * 256 + tid;
    twrd_async_cp16(&t4[e0 < C ? e0 : (C - 1)], &buf[0][tid]);
    for (; c < nChunks; c += gridDim.x) {
      long long cn = c + gridDim.x;            // block-uniform condition
      if (cn < nChunks) {
        long long en = cn * 256 + tid;
        twrd_async_cp16(&t4[en < C ? en : (C - 1)], &buf[p ^ 1][tid]);
        __builtin_amdgcn_s_wait_asynccnt(1);   // oldest copy landed in LDS
      } else {
        __builtin_amdgcn_s_wait_asynccnt(0);
      }
      asm volatile("" ::: "memory");
      long long e = c * 256 + tid;
      if (e < C) {
        float4 vv = buf[p][tid];               // each lane reads its own slot
        m = fmaxf(m, fmaxf(fmaxf(vv.x, vv.y), fmaxf(vv.z, vv.w)));
      }
      p ^= 1;
    }
  }
#else
  for (long long c = blockIdx.x; c < nChunks; c += gridDim.x) {
    long long e = c * 256 + tid;
    if (e < C) {
      float4 vv = t4[e];
      m = fmaxf(m, fmaxf(fmaxf(vv.x, vv.y), fmaxf(vv.z, vv.w)));
    }
  }
#endif

  // scalar tail (E % 4 != 0)
  long long g  = (long long)blockIdx.x * blockDim.x + tid;
  long long gs = (long long)gridDim.x * blockDim.x;
  for (long long e = C * 4 + g; e < E; e += gs) m = fmaxf(m, t[e]);

  // wave32 reduction, then cross-wave via LDS
  __shared__ float wred[8];
  const int lane = tid & 31, wid = tid >> 5;
  for (int o = 16; o > 0; o >>= 1) m = fmaxf(m, __shfl_xor(m, o, 32));
  if (lane == 0) wred[wid] = m;
  __syncthreads();
  if (wid == 0) {
    float x = (lane < 8) ? wred[lane] : 0.0f;
    for (int o = 4; o > 0; o >>= 1) x = fmaxf(x, __shfl_xor(x, o, 32));
    if (lane == 0) atomicMax(maxbits, __float_as_uint(x));
  }
}

// ---------------------------------------------------------------------------
// Kernel 2: deg[u_e] += exp(alpha*(t_e - T))   (f32 global atomics; deg fits
// in L2, so this is L2-atomic bound, not HBM bound)
// ---------------------------------------------------------------------------
__global__ __launch_bounds__(256) void twrd_deg_kernel(const int* __restrict__ u,
                                                       const float* __restrict__ t,
                                                       const unsigned* __restrict__ maxbits,
                                                       float* __restrict__ deg,
                                                       long long E) {
  const float T = __uint_as_float(*maxbits);
  const long long E4 = E >> 2;
  long long g  = (long long)blockIdx.x * blockDim.x + threadIdx.x;
  long long gs = (long long)gridDim.x * blockDim.x;
  const int4*   u4 = (const int4*)u;
  const float4* t4 = (const float4*)t;
  for (long long i = g; i < E4; i += gs) {
    int4 uu = u4[i];
    float4 tt = t4[i];
    atomicAdd(&deg[uu.x], __expf(ALPHA_F * (tt.x - T)));
    atomicAdd(&deg[uu.y], __expf(ALPHA_F * (tt.y - T)));
    atomicAdd(&deg[uu.z], __expf(ALPHA_F * (tt.z - T)));
    atomicAdd(&deg[uu.w], __expf(ALPHA_F * (tt.w - T)));
  }
  for (long long e = E4 * 4 + g; e < E; e += gs)
    atomicAdd(&deg[u[e]], __expf(ALPHA_F * (t[e] - T)));
}

// ---------------------------------------------------------------------------
// Kernel 3: deg -> dinv = deg>0 ? 1/sqrt(deg) : 0   (in place; N from device)
// ---------------------------------------------------------------------------
__global__ __launch_bounds__(256) void twrd_dinv_kernel(float* __restrict__ deg,
                                                        const int* __restrict__ nn) {
  long long N  = *nn;
  long long g  = (long long)blockIdx.x * blockDim.x + threadIdx.x;
  long long gs = (long long)gridDim.x * blockDim.x;
  for (long long i = g; i < N; i += gs) {
    float d = deg[i];
    deg[i] = (d > 0.0f) ? (1.0f / sqrtf(d)) : 0.0f;
  }
}

// ---------------------------------------------------------------------------
// Kernel 4: w_e = dinv[u]*decay*dinv[v]; also emit edge_index as floats.
// Output layout (tuple concat): out[0..E)=u, out[E..2E)=v, out[2E..3E)=w
// ---------------------------------------------------------------------------
__global__ __launch_bounds__(256) void twrd_weight_kernel(const int* __restrict__ u,
                                                          const int* __restrict__ v,
                                                          const float* __restrict__ t,
                                                          const unsigned* __restrict__ maxbits,
                                                          const float* __restrict__ dinv,
                                                          float* __restrict__ out,
                                                          long long E) {
  const float T = __uint_as_float(*maxbits);
  long long g  = (long long)blockIdx.x * blockDim.x + threadIdx.x;
  long long gs = (long long)gridDim.x * blockDim.x;
  const bool vec = ((E & 3) == 0);  // vector stores at out+E/out+2E need E%4==0
  const long long E4 = vec ? (E >> 2) : 0;

  const int4*   u4 = (const int4*)u;
  const int4*   v4 = (const int4*)v;
  const float4* t4 = (const float4*)t;
  float4* ou = (float4*)out;
  float4* ov = (float4*)(out + E);
  float4* ow = (float4*)(out + 2 * E);

  for (long long i = g; i < E4; i += gs) {
    int4 uu = u4[i], vv = v4[i];
    float4 tt = t4[i];
    float4 w;
    w.x = dinv[uu.x] * __expf(ALPHA_F * (tt.x - T)) * dinv[vv.x];
    w.y = dinv[uu.y] * __expf(ALPHA_F * (tt.y - T)) * dinv[vv.y];
    w.z = dinv[uu.z] * __expf(ALPHA_F * (tt.z - T)) * dinv[vv.z];
    w.w = dinv[uu.w] * __expf(ALPHA_F * (tt.w - T)) * dinv[vv.w];
    ou[i] = make_float4((float)uu.x, (float)uu.y, (float)uu.z, (float)uu.w);
    ov[i] = make_float4((float)vv.x, (float)vv.y, (float)vv.z, (float)vv.w);
    ow[i] = w;
  }
  for (long long e = E4 * 4 + g; e < E; e += gs) {
    int ue = u[e], ve = v[e];
    out[e]         = (float)ue;
    out[E + e]     = (float)ve;
    out[2 * E + e] = dinv[ue] * __expf(ALPHA_F * (t[e] - T)) * dinv[ve];
  }
}

// ---------------------------------------------------------------------------
extern "C" void kernel_launch(void* const* d_in, const int* in_sizes, int n_in,
                              void* d_out, int out_size, void* d_ws, size_t ws_size,
                              hipStream_t stream) {
  const int*   edge_index = (const int*)d_in[0];
  const float* edge_time  = (const float*)d_in[1];
  const int*   nn         = (const int*)d_in[2];   // num_nodes (device scalar)
  const long long E = (long long)in_sizes[1];
  const int* u = edge_index;
  const int* v = edge_index + E;

  unsigned* maxbits = (unsigned*)d_ws;
  float*    deg     = (float*)((char*)d_ws + 256);
  float*    out     = (float*)d_out;

  // init (grid-stride handles any N)
  twrd_init_kernel<<<512, 256, 0, stream>>>(maxbits, deg, nn);

  const long long C = E >> 2;
  long long mbll = (C + 255) >> 8;
  if (mbll < 1) mbll = 1;
  if (mbll > 2048) mbll = 2048;
  twrd_max_kernel<<<(int)mbll, 256, 0, stream>>>(edge_time, maxbits, C, E);

  long long dbll = ((E >> 2) + 255) >> 8;
  if (dbll < 1) dbll = 1;
  if (dbll > 4096) dbll = 4096;
  twrd_deg_kernel<<<(int)dbll, 256, 0, stream>>>(u, edge_time, maxbits, deg, E);

  twrd_dinv_kernel<<<512, 256, 0, stream>>>(deg, nn);

  twrd_weight_kernel<<<(int)dbll, 256, 0, stream>>>(u, v, edge_time, maxbits, deg, out, E);
}